// GaussianSplatting_10917806866916
// MI455X (gfx1250) — compile-verified
//
#include <hip/hip_runtime.h>

typedef __attribute__((ext_vector_type(2))) float v2f;
typedef __attribute__((ext_vector_type(8))) float v8f;

#define LOG2E 1.44269504088896340736f
#define EPS2  1e-16f
#define MAX_N_PAD 1032

// One wave32 handles a 16-query tile. Lane (h = lane/16, p = lane%16) computes
// the A-matrix slots (M=p, K=v+2h) for v in {0,1}: i.e. query p against neurons
// n0+2h and n0+2h+1. Three fp32 WMMAs (s, s'*dx/r, s'*dy/r) against B matrices
// holding activities masked into columns 0/1/2 accumulate the (16x3) outputs in
// the 16x16 f32 accumulator for the entire 256-neuron loop.
__global__ __launch_bounds__(256) void GaussianSplatting_kernel(
    const float* __restrict__ positions,   // (N,2)
    const float* __restrict__ activities,  // (N,)
    const float* __restrict__ query,       // (M,2)
    const float* __restrict__ p_ax, const float* __restrict__ p_ay,
    const float* __restrict__ p_tx, const float* __restrict__ p_ty,
    const float* __restrict__ p_sigma, const float* __restrict__ p_beta,
    float* __restrict__ out,               // (M,3)
    int N, int Npad, int M)
{
    __shared__ float4 neur[MAX_N_PAD];     // (px, py, a, 0) per neuron

    const float ax = p_ax[0], ay = p_ay[0];
    const float atx = p_tx[0], aty = p_ty[0];
    const float sigma = p_sigma[0], beta = p_beta[0];

    const int tid = threadIdx.x;

    // Stage affine-transformed neuron data into LDS (4 KB for N=256).
    for (int n = tid; n < Npad; n += 256) {
        float px = 0.0f, py = 0.0f, a = 0.0f;
        if (n < N) {
            px = __builtin_fmaf(ax, positions[2 * n]     - 0.5f, atx + 0.5f);
            py = __builtin_fmaf(ay, positions[2 * n + 1] - 0.5f, aty + 0.5f);
            a  = activities[n];
        }
        neur[n] = make_float4(px, py, a, 0.0f);
    }
    __syncthreads();

    const int lane = tid & 31;
    const int wave = tid >> 5;
    const int h = lane >> 4;               // which half of the wave
    const int p = lane & 15;               // column / A-row index
    const int m_base = blockIdx.x * 128 + wave * 16;

    int mq = m_base + p;
    if (mq >= M) mq = M - 1;               // clamp load; store is predicated
    const float qx = query[2 * mq];
    const float qy = query[2 * mq + 1];

    const float e_scale = beta * LOG2E;            // exp(beta*(r-sigma)) = exp2(fma)
    const float e_bias  = -beta * sigma * LOG2E;
    const float nbeta   = -beta;

    const bool is0 = (p == 0), is1 = (p == 1), is2 = (p == 2);

    v8f acc = {0.f, 0.f, 0.f, 0.f, 0.f, 0.f, 0.f, 0.f};

    for (int n0 = 0; n0 < Npad; n0 += 4) {
        v2f As, Atx, Aty, Bs, Btx, Bty;
        #pragma unroll
        for (int v = 0; v < 2; ++v) {
            const int n = n0 + v + 2 * h;  // A/B slot (vgpr v, half h) <-> K = v+2h
            const float4 nd = neur[n];     // LDS broadcast per half-wave
            const float dx = qx - nd.x;
            const float dy = qy - nd.y;
            const float d2 = __builtin_fmaf(dx, dx, __builtin_fmaf(dy, dy, EPS2));
            const float rinv = __builtin_amdgcn_rsqf(d2);      // v_rsq_f32
            const float r = d2 * rinv;
            const float t = __builtin_amdgcn_exp2f(             // v_exp_f32
                                __builtin_fmaf(r, e_scale, e_bias));
            const float s = __builtin_amdgcn_rcpf(1.0f + t);   // v_rcp_f32
            const float sp = nbeta * s * (1.0f - s);           // t=inf -> s=0 -> sp=0
            const float w = sp * rinv;
            As[v]  = s;
            Atx[v] = w * dx;
            Aty[v] = w * dy;
            const float a = nd.z;
            Bs[v]  = is0 ? a : 0.0f;       // column 0: field
            Btx[v] = is1 ? a : 0.0f;       // column 1: grad_x
            Bty[v] = is2 ? a : 0.0f;       // column 2: grad_y
        }
        acc = __builtin_amdgcn_wmma_f32_16x16x4_f32(false, As,  false, Bs,
                                                    (short)0, acc, false, false);
        acc = __builtin_amdgcn_wmma_f32_16x16x4_f32(false, Atx, false, Btx,
                                                    (short)0, acc, false, false);
        acc = __builtin_amdgcn_wmma_f32_16x16x4_f32(false, Aty, false, Bty,
                                                    (short)0, acc, false, false);
    }

    // D layout: VGPR r, half h -> row M = r + 8h; lane p -> column N = p.
    if (p < 3) {
        #pragma unroll
        for (int r = 0; r < 8; ++r) {
            const int m = m_base + r + 8 * h;
            if (m < M) out[3 * m + p] = acc[r];
        }
    }
}

extern "C" void kernel_launch(void* const* d_in, const int* in_sizes, int n_in,
                              void* d_out, int out_size, void* d_ws, size_t ws_size,
                              hipStream_t stream) {
    (void)n_in; (void)out_size; (void)d_ws; (void)ws_size;
    const float* positions  = (const float*)d_in[0];
    const float* activities = (const float*)d_in[1];
    const float* query      = (const float*)d_in[2];
    const float* p_ax    = (const float*)d_in[3];
    const float* p_ay    = (const float*)d_in[4];
    const float* p_tx    = (const float*)d_in[5];
    const float* p_ty    = (const float*)d_in[6];
    const float* p_sigma = (const float*)d_in[7];
    const float* p_beta  = (const float*)d_in[8];
    float* out = (float*)d_out;

    int N = in_sizes[0] / 2;
    int M = in_sizes[2] / 2;
    int Npad = (N + 3) & ~3;
    if (Npad > MAX_N_PAD) Npad = MAX_N_PAD;   // reference N=256; safety clamp

    const int blocks = (M + 127) / 128;       // 128 queries per 256-thread block
    GaussianSplatting_kernel<<<blocks, 256, 0, stream>>>(
        positions, activities, query,
        p_ax, p_ay, p_tx, p_ty, p_sigma, p_beta,
        out, N, Npad, M);
}